// RGCN_44908178047652
// MI455X (gfx1250) — compile-verified
//
#include <hip/hip_runtime.h>
#include <hip/hip_bf16.h>
#include <math.h>

// ---------------------------------------------------------------------------
// RGCN (2 layers, mean aggregation) restructured as:
//   scatter-add x[src] into agg[(dst,rel)]  ->  fused-scale batched GEMM (WMMA)
// f32 end-to-end via V_WMMA_F32_16X16X4_F32 (memory-bound kernel: the f32
// matrix rate (~1/32 of FP8 peak) still matches the 23.3 TB/s HBM time).
// GEMM register-tiles 80 rows/wave so each B fragment feeds 5 WMMAs, and all
// inner-loop addresses are compile-time immediates (O is a template param).
// ---------------------------------------------------------------------------

typedef float v2f __attribute__((ext_vector_type(2)));
typedef float v8f __attribute__((ext_vector_type(8)));

#define RGCN_N      50000
#define RGCN_E      640000
#define RGCN_R      8
#define RGCN_D      128     // D_IN == D_HID
#define RGCN_MT     5       // 16-row tiles per wave: 80 rows; 50000 = 80*625

// ---------------------------------------------------------------------------
// Scatter: one wave32 per edge; each lane handles 4 contiguous floats.
// agg[(dst*R+rel), 0:128] += feat[src, 0:128]; optional edge count.
// ---------------------------------------------------------------------------
__global__ __launch_bounds__(256)
void rgcn_scatter_kernel(const float* __restrict__ feat,   // [N,128]
                         const int*   __restrict__ esrc,
                         const int*   __restrict__ edst,
                         const int*   __restrict__ etyp,
                         float*       __restrict__ agg,    // [N*R,128]
                         float*       __restrict__ cnt,    // [N*R] or null
                         int n_edges)
{
    const int gid  = blockIdx.x * blockDim.x + threadIdx.x;
    const int e    = gid >> 5;
    const int lane = gid & 31;
    if (e >= n_edges) return;

    const int s = esrc[e];
    const int d = edst[e];
    const int r = etyp[e];
    const long long seg = (long long)d * RGCN_R + r;

    const float4 v = *(const float4*)(feat + (size_t)s * RGCN_D + lane * 4);
    float* p = agg + (size_t)seg * RGCN_D + lane * 4;
    atomicAdd(p + 0, v.x);
    atomicAdd(p + 1, v.y);
    atomicAdd(p + 2, v.z);
    atomicAdd(p + 3, v.w);
    if (cnt != nullptr && lane == 0) atomicAdd(cnt + seg, 1.0f);
}

// inv[i] = 1 / max(cnt[i], 1)
__global__ __launch_bounds__(256)
void rgcn_inv_kernel(const float* __restrict__ cnt, float* __restrict__ inv, int n)
{
    const int i = blockIdx.x * blockDim.x + threadIdx.x;
    if (i < n) inv[i] = 1.0f / fmaxf(cnt[i], 1.0f);
}

// ---------------------------------------------------------------------------
// Fused GEMM:
//   out[n, :] = act( sum_r (agg[n,r,:] * inv[n,r]) @ Wcat[r*128: , :]
//                    + xroot[n,:] @ Wroot + bias )
// Wave owns MT=5 row-tiles (80 rows) x 16 cols. Block = 4 waves = 64 cols.
// A-layout per ISA: lane m=lane&15 holds row m; kh=lane>>4 picks K half.
// ---------------------------------------------------------------------------
template <bool SIGMOID, int O>
__global__ __launch_bounds__(128)
void rgcn_gemm_kernel(const float* __restrict__ agg,    // [N, R*128]
                      const float* __restrict__ inv,    // [N, R]
                      const float* __restrict__ xroot,  // [N, 128]
                      const float* __restrict__ Wcat,   // [R*128, O]
                      const float* __restrict__ Wroot,  // [128, O]
                      const float* __restrict__ bias,   // [O]
                      float*       __restrict__ out)    // [N, O]
{
    const int lane = threadIdx.x & 31;
    const int wave = threadIdx.x >> 5;
    const int col0 = (blockIdx.x * 4 + wave) * 16;
    const int row0 = blockIdx.y * (RGCN_MT * 16);
    const int m    = lane & 15;   // A row / B,C column within tile
    const int kh   = lane >> 4;   // 0: K={k,k+1}, 1: K={k+2,k+3}

    v8f acc[RGCN_MT];
    #pragma unroll
    for (int i = 0; i < RGCN_MT; ++i) acc[i] = (v8f){};

    // Per-row-tile base pointers (even element offset -> 8B-aligned v2f loads).
    const float* abase[RGCN_MT];
    const float* ibase[RGCN_MT];
    const float* xbase[RGCN_MT];
    #pragma unroll
    for (int i = 0; i < RGCN_MT; ++i) {
        const int row = row0 + i * 16 + m;
        abase[i] = agg   + (size_t)row * (RGCN_R * RGCN_D) + 2 * kh;
        ibase[i] = inv   + (size_t)row * RGCN_R;
        xbase[i] = xroot + (size_t)row * RGCN_D + 2 * kh;
    }

    // ---- Relation blocks: K = r*128 .. r*128+127, A scaled by inv[row, r] ----
    for (int r = 0; r < RGCN_R; ++r) {
        float s[RGCN_MT];
        const float* ar[RGCN_MT];
        #pragma unroll
        for (int i = 0; i < RGCN_MT; ++i) {
            s[i]  = ibase[i][r];
            ar[i] = abase[i] + r * RGCN_D;
        }
        const float* bp = Wcat + (size_t)(r * RGCN_D + 2 * kh) * O + col0 + m;

        #pragma unroll
        for (int k = 0; k < RGCN_D; k += 4) {
            v2f b;
            b.x = bp[(size_t)k * O];
            b.y = bp[(size_t)(k + 1) * O];
            #pragma unroll
            for (int i = 0; i < RGCN_MT; ++i) {
                v2f a = *(const v2f*)(ar[i] + k);
                a.x *= s[i];
                a.y *= s[i];
                acc[i] = __builtin_amdgcn_wmma_f32_16x16x4_f32(
                             false, a, false, b, (short)0, acc[i], false, false);
            }
        }
    }

    // ---- Root block (unscaled) ----
    {
        const float* bp = Wroot + (size_t)(2 * kh) * O + col0 + m;
        #pragma unroll
        for (int k = 0; k < RGCN_D; k += 4) {
            v2f b;
            b.x = bp[(size_t)k * O];
            b.y = bp[(size_t)(k + 1) * O];
            #pragma unroll
            for (int i = 0; i < RGCN_MT; ++i) {
                v2f a = *(const v2f*)(xbase[i] + k);
                acc[i] = __builtin_amdgcn_wmma_f32_16x16x4_f32(
                             false, a, false, b, (short)0, acc[i], false, false);
            }
        }
    }

    // ---- Epilogue: C/D layout -> VGPR j holds row (j + 8*kh), col m ----
    const float bv = bias[col0 + m];
    #pragma unroll
    for (int i = 0; i < RGCN_MT; ++i) {
        #pragma unroll
        for (int j = 0; j < 8; ++j) {
            float v = acc[i][j] + bv;
            if (SIGMOID) v = 1.0f / (1.0f + __expf(-v));
            out[(size_t)(row0 + i * 16 + j + 8 * kh) * O + col0 + m] = v;
        }
    }
}

// ---------------------------------------------------------------------------
extern "C" void kernel_launch(void* const* d_in, const int* in_sizes, int n_in,
                              void* d_out, int out_size, void* d_ws, size_t ws_size,
                              hipStream_t stream)
{
    (void)in_sizes; (void)n_in; (void)out_size; (void)ws_size;

    const float* x    = (const float*)d_in[0];
    const int*   esrc = (const int*)  d_in[1];
    const int*   edst = (const int*)  d_in[2];
    const int*   etyp = (const int*)  d_in[3];
    const float* W1   = (const float*)d_in[4];  // [8,128,128] -> [1024,128]
    const float* r1   = (const float*)d_in[5];  // [128,128]
    const float* b1   = (const float*)d_in[6];  // [128]
    const float* W2   = (const float*)d_in[7];  // [8,128,64] -> [1024,64]
    const float* r2   = (const float*)d_in[8];  // [128,64]
    const float* b2   = (const float*)d_in[9];  // [64]

    const int N = RGCN_N, E = RGCN_E, R = RGCN_R, D = RGCN_D;

    // Workspace layout (floats): agg [N*R*128] | cnt [N*R] | inv [N*R] | h [N*128]
    float* agg = (float*)d_ws;
    float* cnt = agg + (size_t)N * R * D;
    float* inv = cnt + (size_t)N * R;
    float* h   = inv + (size_t)N * R;

    const dim3 sblk(256);
    const dim3 sgrd((unsigned)(((size_t)E * 32 + 255) / 256));
    const dim3 gblk(128);
    const unsigned gy = N / (16 * RGCN_MT);   // 625

    // ---- Layer 1 ----
    hipMemsetAsync(agg, 0, (size_t)N * R * D * sizeof(float), stream);
    hipMemsetAsync(cnt, 0, (size_t)N * R * sizeof(float), stream);
    rgcn_scatter_kernel<<<sgrd, sblk, 0, stream>>>(x, esrc, edst, etyp, agg, cnt, E);
    rgcn_inv_kernel<<<(N * R + 255) / 256, 256, 0, stream>>>(cnt, inv, N * R);
    rgcn_gemm_kernel<false, 128><<<dim3(2, gy), gblk, 0, stream>>>(
        agg, inv, x, W1, r1, b1, h);

    // ---- Layer 2 (same edge counts -> reuse inv) ----
    hipMemsetAsync(agg, 0, (size_t)N * R * D * sizeof(float), stream);
    rgcn_scatter_kernel<<<sgrd, sblk, 0, stream>>>(h, esrc, edst, etyp, agg, nullptr, E);
    rgcn_gemm_kernel<true, 64><<<dim3(1, gy), gblk, 0, stream>>>(
        agg, inv, h, W2, r2, b2, (float*)d_out);
}